// DeepGESNCell_81037442941024
// MI455X (gfx1250) — compile-verified
//
#include <hip/hip_runtime.h>
#include <math.h>

// ---------------------------------------------------------------------------
// DeepGESN on MI455X (gfx1250), fp32 end-to-end via V_WMMA_F32_16X16X4_F32.
//   layer(x): wiu = x @ W_ih^T ; h=0; 10x: h = tanh(wiu + L @ (h @ W_hh^T))
//   out = concat(h1, h2)
// L (64 MB) is L2-resident (192 MB L2); big GEMM is matrix-core bound.
// g is kept TRANSPOSED (gT[64][4096]) so B tiles are contiguous float2 along
// K; M=32 per wave + K-split partials give 6 b64 loads per 8 WMMAs and up to
// 1024 concurrent waves (deterministic: per-segment buffers, no atomics).
// ---------------------------------------------------------------------------

#define NN 4096
#define HH 64

typedef __attribute__((ext_vector_type(2))) float v2f;
typedef __attribute__((ext_vector_type(8))) float v8f;

static __device__ __forceinline__ v8f wmma_f32(v2f a, v2f b, v8f c) {
  // 8 args: (neg_a, A, neg_b, B, c_mod, C, reuse_a, reuse_b)
  return __builtin_amdgcn_wmma_f32_16x16x4_f32(false, a, false, b,
                                               (short)0, c, false, false);
}

// OUT = IN[NN,HH] @ W[HH,HH]^T.
// TRANS=0: OUT[NN,HH] row-major (wiu).  TRANS=1: OUT^T[HH,NN] (gT for big GEMM).
template <bool TRANS>
__global__ __launch_bounds__(32)
void gesn_small(const float* __restrict__ IN, const float* __restrict__ W,
                float* __restrict__ OUT) {
  const int lane = threadIdx.x & 31;
  const int l15  = lane & 15;
  const int up   = lane >> 4;            // 0: lanes 0-15, 1: lanes 16-31
  const int m0   = blockIdx.x * 16;

  v8f acc[4] = {};
  const float* Ap = IN + (size_t)(m0 + l15) * HH + 2 * up;

#pragma unroll
  for (int k0 = 0; k0 < HH; k0 += 4) {
    v2f a = *(const v2f*)(Ap + k0);      // A: rows of IN
    const int kr = k0 + 2 * up;
#pragma unroll
    for (int t = 0; t < 4; ++t) {
      // B[k][n] = W^T[k][n] = W[n][k] -> contiguous float2 along k
      v2f b = *(const v2f*)(W + (size_t)(16 * t + l15) * HH + kr);
      acc[t] = wmma_f32(a, b, acc[t]);
    }
  }

#pragma unroll
  for (int t = 0; t < 4; ++t) {
    const int col  = 16 * t + l15;
    const int row0 = m0 + 8 * up;        // acc[t][v] is row row0+v, same col
    if (TRANS) {
      // 8 consecutive rows at fixed col are contiguous in OUT^T: one v8f store
      *(v8f*)(OUT + (size_t)col * NN + row0) = acc[t];
    } else {
#pragma unroll
      for (int v = 0; v < 8; ++v)
        OUT[(size_t)(row0 + v) * HH + col] = acc[t][v];
    }
  }
}

// part[seg] = L[m0:m0+32, kseg] @ gT[:, kseg]^T   (K-split partial GEMM)
// grid: (NN/32, S); each wave: 32x64 output tile over klen K values.
__global__ __launch_bounds__(32)
void gesn_big_part(const float* __restrict__ L, const float* __restrict__ gT,
                   float* __restrict__ part, int klen) {
  const int lane  = threadIdx.x & 31;
  const int l15   = lane & 15;
  const int up    = lane >> 4;
  const int m0    = blockIdx.x * 32;
  const int kbase = blockIdx.y * klen;
  float* p = part + (size_t)blockIdx.y * NN * HH;

  v8f acc[2][4] = {};
  const float* Lp0 = L  + (size_t)(m0 + l15) * NN + kbase + 2 * up;
  const float* Lp1 = Lp0 + (size_t)16 * NN;
  const float* Bp  = gT + (size_t)l15 * NN + kbase + 2 * up;  // + t*16*NN + k0

#pragma unroll 4
  for (int k0 = 0; k0 < klen; k0 += 4) {
    v2f a0 = *(const v2f*)(Lp0 + k0);
    v2f a1 = *(const v2f*)(Lp1 + k0);
#pragma unroll
    for (int t = 0; t < 4; ++t) {
      v2f b = *(const v2f*)(Bp + (size_t)t * 16 * NN + k0);  // gT[n][kr,kr+1]
      acc[0][t] = wmma_f32(a0, b, acc[0][t]);
      acc[1][t] = wmma_f32(a1, b, acc[1][t]);
    }
  }

#pragma unroll
  for (int mh = 0; mh < 2; ++mh)
#pragma unroll
    for (int t = 0; t < 4; ++t)
#pragma unroll
      for (int v = 0; v < 8; ++v) {
        const int row = m0 + 16 * mh + 8 * up + v;
        const int col = 16 * t + l15;
        p[(size_t)row * HH + col] = acc[mh][t][v];
      }
}

// h = tanh(wiu + sum_s part[s])
__global__ void gesn_fix(const float* __restrict__ wiu,
                         const float* __restrict__ part, int S,
                         float* __restrict__ h) {
  const int i = blockIdx.x * blockDim.x + threadIdx.x;
  if (i < NN * HH) {
    float z = wiu[i];
    for (int s = 0; s < S; ++s) z += part[(size_t)s * NN * HH + i];
    h[i] = tanhf(z);
  }
}

// h = tanh(wiu)   (iteration 1: h0 == 0 so L@(h0 W^T) == 0)
__global__ void gesn_tanh(const float* __restrict__ wiu, float* __restrict__ h,
                          int n) {
  const int i = blockIdx.x * blockDim.x + threadIdx.x;
  if (i < n) h[i] = tanhf(wiu[i]);
}

// out[:, colofs:colofs+64] = h
__global__ void gesn_copy(const float* __restrict__ h, float* __restrict__ out,
                          int colofs) {
  const int i = blockIdx.x * blockDim.x + threadIdx.x;
  if (i < NN * HH) {
    const int r = i >> 6, c = i & 63;
    out[(size_t)r * (2 * HH) + colofs + c] = h[i];
  }
}

extern "C" void kernel_launch(void* const* d_in, const int* in_sizes, int n_in,
                              void* d_out, int out_size, void* d_ws,
                              size_t ws_size, hipStream_t stream) {
  (void)in_sizes; (void)n_in; (void)out_size;
  const float* X    = (const float*)d_in[0];
  const float* L    = (const float*)d_in[1];
  const float* Wih0 = (const float*)d_in[2];
  const float* Whh0 = (const float*)d_in[3];
  const float* Wih1 = (const float*)d_in[4];
  const float* Whh1 = (const float*)d_in[5];
  float* out = (float*)d_out;

  const size_t slab = (size_t)NN * HH;   // 262144 floats = 1 MB
  float* wiu  = (float*)d_ws;            // [slab]
  float* h    = wiu + slab;              // [slab]
  float* gT   = h + slab;                // [slab]  (transposed: [HH][NN])
  float* part = gT + slab;               // [S * slab]

  // K-split factor from available workspace (deterministic for a given run).
  const size_t avail = ws_size / sizeof(float);
  int S = 1;
  if (avail > 4 * slab) {
    S = (int)((avail - 3 * slab) / slab);
    if (S > 8) S = 8;
  }
  while (S & (S - 1)) --S;               // power of two: 1,2,4,8
  const int klen = NN / S;

  const dim3 gs(NN / 16), gbig(NN / 32, S), wv(32);
  const int NE = NN * HH;
  const dim3 eg((NE + 255) / 256), ebk(256);

  // ----- layer 0 -----
  gesn_small<false><<<gs, wv, 0, stream>>>(X, Wih0, wiu);
  gesn_tanh<<<eg, ebk, 0, stream>>>(wiu, h, NE);
  for (int it = 0; it < 9; ++it) {
    gesn_small<true><<<gs, wv, 0, stream>>>(h, Whh0, gT);
    gesn_big_part<<<gbig, wv, 0, stream>>>(L, gT, part, klen);
    gesn_fix<<<eg, ebk, 0, stream>>>(wiu, part, S, h);
  }
  gesn_copy<<<eg, ebk, 0, stream>>>(h, out, 0);        // h1 -> out[:, :64]

  // ----- layer 1 (input = h1, still live in h) -----
  gesn_small<false><<<gs, wv, 0, stream>>>(h, Wih1, wiu);  // before h overwrite
  gesn_tanh<<<eg, ebk, 0, stream>>>(wiu, h, NE);
  for (int it = 0; it < 9; ++it) {
    gesn_small<true><<<gs, wv, 0, stream>>>(h, Whh1, gT);
    gesn_big_part<<<gbig, wv, 0, stream>>>(L, gT, part, klen);
    gesn_fix<<<eg, ebk, 0, stream>>>(wiu, part, S, h);
  }
  gesn_copy<<<eg, ebk, 0, stream>>>(h, out, 64);       // h2 -> out[:, 64:]
}